// KernelCRPS_47777216200759
// MI455X (gfx1250) — compile-verified
//
#include <hip/hip_runtime.h>

typedef __attribute__((ext_vector_type(2))) float v2f;
typedef __attribute__((ext_vector_type(4))) float v4f;
typedef __attribute__((ext_vector_type(8))) float v8f;

namespace {
constexpr int kBatch   = 2;
constexpr int kNVars   = 16;
constexpr int kLatLon  = 40320;
constexpr int kEns     = 16;
constexpr int kNPoints = kBatch * kNVars * kLatLon;   // 1,290,240
constexpr int kBlocks  = 1024;                         // partials: 1024 f32 = 4 KB ws
constexpr int kThreads = 256;                          // 8 waves (wave32)
constexpr int kWaves   = kThreads / 32;
}

// Kernel 1: streaming CRPS partial sums. One point (b,v,p) per thread per
// grid-stride iteration: 4x b128 NT loads of the 16-member ensemble, register
// pairwise |diff| reduction, deterministic wave+block reduce -> ws[block].
__global__ __launch_bounds__(kThreads) void crps_partial(
    const float* __restrict__ y_pred,
    const float* __restrict__ y_target,
    const float* __restrict__ weights,
    const float* __restrict__ scale,
    float* __restrict__ block_partials) {
  const int tid = threadIdx.x;
  const int stride = gridDim.x * blockDim.x;
  float acc = 0.0f;

  for (int idx = blockIdx.x * blockDim.x + tid; idx < kNPoints; idx += stride) {
    const v4f* pp = reinterpret_cast<const v4f*>(y_pred) + (size_t)idx * (kEns / 4);
    // Speculative prefetch of the next grid-stride point (global_prefetch_b8).
    __builtin_prefetch(pp + (size_t)stride * (kEns / 4), 0, 0);
    // Single-pass stream: non-temporal 128-bit loads.
    v4f q0 = __builtin_nontemporal_load(pp + 0);
    v4f q1 = __builtin_nontemporal_load(pp + 1);
    v4f q2 = __builtin_nontemporal_load(pp + 2);
    v4f q3 = __builtin_nontemporal_load(pp + 3);

    float p[kEns];
#pragma unroll
    for (int e = 0; e < 4; ++e) {
      p[e] = q0[e]; p[4 + e] = q1[e]; p[8 + e] = q2[e]; p[12 + e] = q3[e];
    }

    const float t = y_target[idx];
    float mae = 0.0f;
#pragma unroll
    for (int e = 0; e < kEns; ++e) mae += __builtin_fabsf(t - p[e]);

    float pair = 0.0f;  // sum_{i<j} |p_i - p_j|; full {i,j} sum = 2*pair
#pragma unroll
    for (int i = 0; i < kEns; ++i) {
#pragma unroll
      for (int j = i + 1; j < kEns; ++j) pair += __builtin_fabsf(p[i] - p[j]);
    }

    const int v  = (idx / kLatLon) % kNVars;
    const int pt = idx % kLatLon;
    // mae/E + coef*2*pair, coef = -1/(2E^2)  ->  mae/E - pair/E^2
    const float val = mae * (1.0f / kEns) - pair * (1.0f / (kEns * kEns));
    acc += val * (scale[v] * weights[pt]);
  }

  // wave32 butterfly reduce (deterministic)
#pragma unroll
  for (int off = 16; off > 0; off >>= 1) acc += __shfl_xor(acc, off, 32);

  __shared__ float sdata[kWaves];
  if ((tid & 31) == 0) sdata[tid >> 5] = acc;
  __syncthreads();
  if (tid == 0) {
    float s = 0.0f;
#pragma unroll
    for (int w = 0; w < kWaves; ++w) s += sdata[w];
    block_partials[blockIdx.x] = s;
  }
}

// Kernel 2: weight sum + WMMA-based deterministic reduction of the 1024 block
// partials (V_WMMA_F32_16X16X4_F32 with B = ones: row-sums over K), then the
// final normalization: out = total / (sum(weights) * BATCH).
__global__ __launch_bounds__(kThreads) void crps_final(
    const float* __restrict__ block_partials,
    const float* __restrict__ weights,
    float* __restrict__ out) {
  const int tid = threadIdx.x;
  __shared__ float sdata[kWaves];
  __shared__ float wsum_sh;

  // ---- sum(weights) over LATLON ----
  float w = 0.0f;
  for (int i = tid; i < kLatLon; i += kThreads) w += weights[i];
#pragma unroll
  for (int off = 16; off > 0; off >>= 1) w += __shfl_xor(w, off, 32);
  if ((tid & 31) == 0) sdata[tid >> 5] = w;
  __syncthreads();
  if (tid == 0) {
    float s = 0.0f;
#pragma unroll
    for (int k = 0; k < kWaves; ++k) s += sdata[k];
    wsum_sh = s;
  }
  __syncthreads();

  // ---- WMMA reduce of kBlocks partials on wave 0 (EXEC all-ones) ----
  if (tid < 32) {
    const int lane = tid;
    v8f cacc = {};              // 16x16 f32 accumulator, starts at 0
    v2f bones;                  // B = ones(4x16): every K,N element is 1.0
    bones.x = 1.0f; bones.y = 1.0f;
    // A 16x4 f32 layout: lanes 0-15 hold (m=lane, k=0/1), lanes 16-31 hold
    // (m=lane-16, k=2/3). Branchless index keeps EXEC uniform for WMMA.
    const int aidx = (lane & 15) * 4 + ((lane >> 4) << 1);
#pragma unroll
    for (int c = 0; c < kBlocks / 64; ++c) {   // 16 chunks of 64 partials
      const float* base = block_partials + c * 64;
      v2f a;
      a.x = base[aidx];
      a.y = base[aidx + 1];
      // D = A x ones + C : every column of D is the K-rowsum of A.
      cacc = __builtin_amdgcn_wmma_f32_16x16x4_f32(
          /*neg_a=*/false, a, /*neg_b=*/false, bones,
          /*c_mod=*/(short)0, cacc, /*reuse_a=*/false, /*reuse_b=*/false);
    }
    // Column 0 of C: lane 0 holds C[0..7,0] in cacc[0..7], lane 16 holds
    // C[8..15,0]. Total = sum of those 16 values.
    float s = cacc[0] + cacc[1] + cacc[2] + cacc[3] +
              cacc[4] + cacc[5] + cacc[6] + cacc[7];
    const float hi = __shfl(s, 16, 32);
    if (lane == 0) {
      out[0] = (s + hi) / (wsum_sh * (float)kBatch);
    }
  }
}

extern "C" void kernel_launch(void* const* d_in, const int* in_sizes, int n_in,
                              void* d_out, int out_size, void* d_ws, size_t ws_size,
                              hipStream_t stream) {
  const float* y_pred   = (const float*)d_in[0];
  const float* y_target = (const float*)d_in[1];
  const float* weights  = (const float*)d_in[2];
  const float* scale    = (const float*)d_in[3];
  float* partials = (float*)d_ws;   // kBlocks floats = 4 KB
  float* out      = (float*)d_out;

  crps_partial<<<kBlocks, kThreads, 0, stream>>>(y_pred, y_target, weights,
                                                 scale, partials);
  crps_final<<<1, kThreads, 0, stream>>>(partials, weights, out);

  (void)in_sizes; (void)n_in; (void)out_size; (void)ws_size;
}